// ExtAttentionPool_86672440033956
// MI455X (gfx1250) — compile-verified
//
#include <hip/hip_runtime.h>

// CDNA5 WMMA vector types (wave32): f32 16x16x4 -> A,B = 2 VGPRs/lane, C/D = 8 VGPRs/lane
typedef __attribute__((ext_vector_type(2))) float v2f;
typedef __attribute__((ext_vector_type(8))) float v8f;

#define B_  16
#define T_  1024
#define D_  1024
#define O_  10
#define OP_ 16   // O padded to WMMA tile width

// -----------------------------------------------------------------------------
// Kernel 1: scores_t[b][o][t] = (dot(logits[b,t,:], W[o,:]) + bias[o]) / O
// One wave per 16(T-rows) x 16(O-cols) tile; K-loop of V_WMMA_F32_16X16X4_F32.
// W is staged zero-padded into LDS so padded columns contribute exact zeros.
// -----------------------------------------------------------------------------
__global__ __launch_bounds__(256) void k_scores(const float* __restrict__ logits,
                                                const float* __restrict__ W,
                                                const float* __restrict__ bias,
                                                float* __restrict__ scores) {
  __shared__ float Wlds[OP_ * D_];          // 64 KB of the 320 KB WGP LDS
  const int tid = threadIdx.x;
  for (int idx = tid; idx < OP_ * D_; idx += 256) {
    const int row = idx >> 10;              // padded O row
    const int col = idx & (D_ - 1);
    Wlds[idx] = (row < O_) ? W[row * D_ + col] : 0.0f;
  }
  __syncthreads();

  const int wave = tid >> 5;
  const int lane = tid & 31;
  const int gw   = blockIdx.x * 8 + wave;   // 128 blocks * 8 waves = 1024 tiles
  const int b    = gw >> 6;                 // 64 T-tiles per batch
  const int t0   = (gw & 63) << 4;
  const int half = lane >> 4;
  const int l15  = lane & 15;
  const int koff = half * 2;                // A/B k-pair per lane half

  // A: lane holds row M=l15, k-pair (koff, koff+1) per ISA 16x4 f32 layout
  const float* aRow = logits + ((size_t)b * T_ + (t0 + l15)) * D_ + koff;
  // B: lane holds col N=l15 of W^T; Wlds row-major (n, k) -> contiguous k pair
  const float* bRow = Wlds + l15 * D_ + koff;

  v8f c = {};
  #pragma unroll 4
  for (int k = 0; k < D_; k += 4) {
    v2f a  = *reinterpret_cast<const v2f*>(aRow + k);   // global b64
    v2f bb = *reinterpret_cast<const v2f*>(bRow + k);   // ds_load_b64
    c = __builtin_amdgcn_wmma_f32_16x16x4_f32(false, a, false, bb,
                                              (short)0, c, false, false);
  }

  const float bn  = (l15 < O_) ? bias[l15] : 0.0f;
  float* dst = scores + ((size_t)b * OP_ + l15) * T_ + t0;  // (B,16,T) layout
  #pragma unroll
  for (int v = 0; v < 8; ++v) {
    const int m = v + 8 * half;             // C/D layout: VGPR v -> M = v + 8*half
    dst[m] = (c[v] + bn) * (1.0f / O_);
  }
}

// -----------------------------------------------------------------------------
// Kernel 2: in-place softmax over the contiguous T axis of each (b, o_pad) row.
// 256 rows x 1024 elements; one 256-thread block per row, 4 elems/thread.
// -----------------------------------------------------------------------------
__global__ __launch_bounds__(256) void k_softmax(float* __restrict__ ws) {
  __shared__ float red[256];
  float* p = ws + (size_t)blockIdx.x * T_;
  const int tid = threadIdx.x;

  float x0 = p[tid], x1 = p[tid + 256], x2 = p[tid + 512], x3 = p[tid + 768];
  red[tid] = fmaxf(fmaxf(x0, x1), fmaxf(x2, x3));
  __syncthreads();
  for (int s = 128; s > 0; s >>= 1) {
    if (tid < s) red[tid] = fmaxf(red[tid], red[tid + s]);
    __syncthreads();
  }
  const float m = red[0];
  __syncthreads();

  const float e0 = __expf(x0 - m), e1 = __expf(x1 - m);
  const float e2 = __expf(x2 - m), e3 = __expf(x3 - m);
  red[tid] = e0 + e1 + e2 + e3;
  __syncthreads();
  for (int s = 128; s > 0; s >>= 1) {
    if (tid < s) red[tid] += red[tid + s];
    __syncthreads();
  }
  const float inv = 1.0f / red[0];
  p[tid] = e0 * inv; p[tid + 256] = e1 * inv;
  p[tid + 512] = e2 * inv; p[tid + 768] = e3 * inv;
}

// -----------------------------------------------------------------------------
// Kernel 3: out[b, o*T + t] = sum_i logits[b,t,i] * w_x[b,i,o]
// Same WMMA tiling; B-operand read from transposed w_x (B,16,T) -> contiguous
// k-pairs per lane. Store predicated to the 10 real output columns.
// -----------------------------------------------------------------------------
__global__ __launch_bounds__(256) void k_out(const float* __restrict__ logits,
                                             const float* __restrict__ wx,
                                             float* __restrict__ out) {
  const int tid  = threadIdx.x;
  const int wave = tid >> 5;
  const int lane = tid & 31;
  const int gw   = blockIdx.x * 8 + wave;
  const int b    = gw >> 6;
  const int t0   = (gw & 63) << 4;
  const int half = lane >> 4;
  const int l15  = lane & 15;
  const int koff = half * 2;

  const float* aRow = logits + ((size_t)b * T_ + (t0 + l15)) * D_ + koff;
  const float* bRow = wx + ((size_t)b * OP_ + l15) * T_ + koff;

  v8f c = {};
  #pragma unroll 4
  for (int k = 0; k < D_; k += 4) {
    v2f a  = *reinterpret_cast<const v2f*>(aRow + k);
    v2f bb = *reinterpret_cast<const v2f*>(bRow + k);
    c = __builtin_amdgcn_wmma_f32_16x16x4_f32(false, a, false, bb,
                                              (short)0, c, false, false);
  }

  if (l15 < O_) {                            // drop padded O columns
    float* dst = out + (size_t)b * (O_ * T_) + l15 * T_ + t0;
    #pragma unroll
    for (int v = 0; v < 8; ++v) dst[v + 8 * half] = c[v];
  }
}

// -----------------------------------------------------------------------------
extern "C" void kernel_launch(void* const* d_in, const int* in_sizes, int n_in,
                              void* d_out, int out_size, void* d_ws, size_t ws_size,
                              hipStream_t stream) {
  const float* logits = (const float*)d_in[0];
  // d_in[1] = decision (unused by forward math)
  const float* W    = (const float*)d_in[2];
  const float* bias = (const float*)d_in[3];
  float*       out  = (float*)d_out;
  float*       ws   = (float*)d_ws;          // B*16*T floats = 1 MB scratch

  // GEMM1 + bias + /O  -> scores in (B,16,T) layout
  k_scores<<<128, 256, 0, stream>>>(logits, W, bias, ws);
  // softmax over T, in place (scores -> w_x)
  k_softmax<<<B_ * OP_, 256, 0, stream>>>(ws);
  // GEMM2 -> (B, O*T) output; logits re-read should hit the 192 MB L2
  k_out<<<128, 256, 0, stream>>>(logits, ws, out);
}